// Attention_35605278884484
// MI455X (gfx1250) — compile-verified
//
#include <hip/hip_runtime.h>
#include <math.h>

// Problem constants (reference: H=8, DH=64, B=1, N=512, DN=128, DE=64)
constexpr int N_     = 512;
constexpr int DNODE  = 128;
constexpr int DE_    = 64;
constexpr int H_     = 8;
constexpr int DH_    = 64;
constexpr int INNER_ = 512;
constexpr float SCALE = 0.125f;          // DH^-0.5
constexpr int EDS = DE_ + 1;             // padded LDS stride for edges[i] tile (bank-conflict-free)
constexpr int SS  = N_ + 1;              // padded LDS stride for sim/attn rows

typedef float v2f __attribute__((ext_vector_type(2)));
typedef float v8f __attribute__((ext_vector_type(8)));

// D(16x16,f32) = A(16x4,f32) * B(4x16,f32) + C  -- wave32, one tile per wave.
// A: lane l holds A[l&15][2*(l>>4)+g] in vgpr g; B: lane l holds B[2*(l>>4)+g][l&15];
// C/D: vgpr r, lane l -> D[r + 8*(l>>4)][l&15].
__device__ __forceinline__ v8f wmma4(v2f a, v2f b, v8f c) {
  return __builtin_amdgcn_wmma_f32_16x16x4_f32(false, a, false, b, (short)0, c,
                                               false, false);
}

// ---------------------------------------------------------------------------
// 1) Q/K/V projections.  Q = nodes@Wq + bq ; K = nodes@Wk + bk + be ;
//    V = nodes@Wv + bv + be   (edge bias be folds exactly into K,V since
//    k_ij = k_j + e_ij, v_ij = v_j + e_ij and softmax rows sum to 1).
//    Output layout: [h][n][dh].
// ---------------------------------------------------------------------------
__global__ __launch_bounds__(32) void k_qkv(
    const float* __restrict__ nodes,
    const float* __restrict__ Wq, const float* __restrict__ bq,
    const float* __restrict__ Wk, const float* __restrict__ bk,
    const float* __restrict__ Wv, const float* __restrict__ bv,
    const float* __restrict__ be,
    float* __restrict__ Q, float* __restrict__ K, float* __restrict__ V)
{
  const int lane = threadIdx.x & 31;
  const int mrow = lane & 15, khalf = lane >> 4;
  const int col0 = blockIdx.x * 16, row0 = blockIdx.y * 16, mat = blockIdx.z;
  const float* W   = (mat == 0) ? Wq : (mat == 1) ? Wk : Wv;
  const float* bs  = (mat == 0) ? bq : (mat == 1) ? bk : bv;
  float*       Dst = (mat == 0) ? Q  : (mat == 1) ? K  : V;

  v8f acc = {};
  for (int kk = 0; kk < DNODE; kk += 4) {
    const int ka = kk + 2 * khalf;
    v2f a, b;
    a.x = nodes[(row0 + mrow) * DNODE + ka];
    a.y = nodes[(row0 + mrow) * DNODE + ka + 1];
    b.x = W[ka * INNER_ + col0 + mrow];
    b.y = W[(ka + 1) * INNER_ + col0 + mrow];
    acc = wmma4(a, b, acc);
  }
  const int col = col0 + mrow;
  const float bias = bs[col] + ((mat != 0) ? be[col] : 0.0f);
  const int h = col >> 6, d = col & 63;
  for (int r = 0; r < 8; ++r) {
    const int n = row0 + r + 8 * khalf;
    Dst[(h * N_ + n) * DH_ + d] = acc[r] + bias;
  }
}

// ---------------------------------------------------------------------------
// 2) QE[h,n,c] = sum_d Q[h,n,d] * We[c, h*64+d]   (query projected into edge
//    feature space; makes the per-(i,j) edge-key dot a GEMM against edges[i]).
// ---------------------------------------------------------------------------
__global__ __launch_bounds__(32) void k_qe(
    const float* __restrict__ Q, const float* __restrict__ We,
    float* __restrict__ QE)
{
  const int lane = threadIdx.x & 31;
  const int mrow = lane & 15, khalf = lane >> 4;
  const int col0 = blockIdx.x * 16, row0 = blockIdx.y * 16, h = blockIdx.z;

  v8f acc = {};
  for (int kk = 0; kk < DH_; kk += 4) {
    const int ka = kk + 2 * khalf;
    v2f a, b;
    a.x = Q[(h * N_ + row0 + mrow) * DH_ + ka];
    a.y = Q[(h * N_ + row0 + mrow) * DH_ + ka + 1];
    b.x = We[(col0 + mrow) * INNER_ + h * DH_ + ka];
    b.y = We[(col0 + mrow) * INNER_ + h * DH_ + ka + 1];
    acc = wmma4(a, b, acc);
  }
  for (int r = 0; r < 8; ++r) {
    const int n = row0 + r + 8 * khalf;
    QE[(h * N_ + n) * DE_ + col0 + mrow] = acc[r];
  }
}

// ---------------------------------------------------------------------------
// 3) SQK[h,i,j] = Q[h,i,:] . K[h,j,:]   (per-head Q K^T, K already holds +be)
// ---------------------------------------------------------------------------
__global__ __launch_bounds__(32) void k_sqk(
    const float* __restrict__ Q, const float* __restrict__ K,
    float* __restrict__ SQK)
{
  const int lane = threadIdx.x & 31;
  const int mrow = lane & 15, khalf = lane >> 4;
  const int j0 = blockIdx.x * 16, i0 = blockIdx.y * 16, h = blockIdx.z;

  v8f acc = {};
  for (int kk = 0; kk < DH_; kk += 4) {
    const int ka = kk + 2 * khalf;
    v2f a, b;
    a.x = Q[(h * N_ + i0 + mrow) * DH_ + ka];
    a.y = Q[(h * N_ + i0 + mrow) * DH_ + ka + 1];
    b.x = K[(h * N_ + j0 + mrow) * DH_ + ka];
    b.y = K[(h * N_ + j0 + mrow) * DH_ + ka + 1];
    acc = wmma4(a, b, acc);
  }
  for (int r = 0; r < 8; ++r) {
    const int i = i0 + r + 8 * khalf;
    SQK[((size_t)h * N_ + i) * N_ + j0 + mrow] = acc[r];
  }
}

// ---------------------------------------------------------------------------
// 4) Attention core: one workgroup (8 waves) per query row i.
//    edges[i] (512x64 f32 = 128KB) staged once in dynamic LDS (padded stride),
//    used for BOTH the edge-sim GEMM and the attention-weighted edge sum.
// ---------------------------------------------------------------------------
__global__ __launch_bounds__(256) void k_attn(
    const float* __restrict__ edges, const unsigned char* __restrict__ mask,
    const float* __restrict__ QE, const float* __restrict__ SQK,
    float* __restrict__ ATTN, float* __restrict__ WEsum)
{
  extern __shared__ float smem[];
  float* eds  = smem;               // [N_][EDS]  edges[i], padded stride
  float* sims = eds + N_ * EDS;     // [H_][SS]   sim -> attn in place
  float* qes  = sims + H_ * SS;     // [H_][DE_]  QE rows for this i

  const int tid = threadIdx.x, wave = tid >> 5, lane = tid & 31;
  const int mrow = lane & 15, khalf = lane >> 4;
  const int i = blockIdx.x;
  const float* eg = edges + (size_t)i * N_ * DE_;

  // Prefetch next i's 128KB edge slab while we work on this one.
  if (i + 1 < N_) {
    const char* nxt = (const char*)(edges + (size_t)(i + 1) * N_ * DE_);
    __builtin_prefetch(nxt + tid * 512, 0, 1);
  }

  for (int t = tid; t < N_ * DE_; t += 256)
    eds[(t >> 6) * EDS + (t & 63)] = eg[t];
  for (int t = tid; t < H_ * DE_; t += 256)
    qes[t] = QE[((t >> 6) * N_ + i) * DE_ + (t & 63)];
  __syncthreads();

  const bool mi = (mask[i] != 0);

  // sim[h][j] = SCALE * (SQK[h,i,j] + sum_c QE[h,i,c]*edges[i,j,c])
  // A = QE rows (heads 0..7, rows 8..15 zero), B = edges[i]^T from LDS.
  for (int jt = wave; jt < N_ / 16; jt += 8) {
    const int j0 = jt * 16;
    v8f acc = {};
    for (int kk = 0; kk < DE_; kk += 4) {
      const int ka = kk + 2 * khalf;
      v2f a, b;
      a.x = (mrow < 8) ? qes[mrow * DE_ + ka]     : 0.0f;
      a.y = (mrow < 8) ? qes[mrow * DE_ + ka + 1] : 0.0f;
      b.x = eds[(j0 + mrow) * EDS + ka];
      b.y = eds[(j0 + mrow) * EDS + ka + 1];
      acc = wmma4(a, b, acc);
    }
    if (khalf == 0) {  // rows 0..7 carry the 8 heads
      const int j = j0 + mrow;
      const bool ok = mi && (mask[j] != 0);
      for (int r = 0; r < 8; ++r) {
        const float s = (acc[r] + SQK[((size_t)r * N_ + i) * N_ + j]) * SCALE;
        sims[r * SS + j] = ok ? s : -3.402823466e38f;
      }
    }
  }
  __syncthreads();

  // Softmax over j: wave w owns head w (wave32 shfl_xor tree reductions).
  {
    const int h = wave;
    float mx = -3.402823466e38f;
    for (int j = lane; j < N_; j += 32) mx = fmaxf(mx, sims[h * SS + j]);
    for (int o = 16; o; o >>= 1) mx = fmaxf(mx, __shfl_xor(mx, o, 32));
    float sum = 0.0f;
    for (int j = lane; j < N_; j += 32) {
      const float e = __expf(sims[h * SS + j] - mx);
      sims[h * SS + j] = e;
      sum += e;
    }
    for (int o = 16; o; o >>= 1) sum += __shfl_xor(sum, o, 32);
    const float inv = 1.0f / sum;
    for (int j = lane; j < N_; j += 32) {
      const float a = sims[h * SS + j] * inv;
      sims[h * SS + j] = a;
      ATTN[((size_t)h * N_ + i) * N_ + j] = a;
    }
  }
  __syncthreads();

  // WEsum[h,i,c] = sum_j attn[h,i,j] * edges[i,j,c]
  // A = attn rows (heads), B = edges[i] from LDS; waves 0..3 take one c-tile.
  if (wave < 4) {
    const int c0 = wave * 16;
    v8f acc = {};
    for (int kk = 0; kk < N_; kk += 4) {
      const int ka = kk + 2 * khalf;
      v2f a, b;
      a.x = (mrow < 8) ? sims[mrow * SS + ka]     : 0.0f;
      a.y = (mrow < 8) ? sims[mrow * SS + ka + 1] : 0.0f;
      b.x = eds[ka * EDS + c0 + mrow];
      b.y = eds[(ka + 1) * EDS + c0 + mrow];
      acc = wmma4(a, b, acc);
    }
    if (khalf == 0) {
      for (int r = 0; r < 8; ++r)
        WEsum[((size_t)r * N_ + i) * DE_ + c0 + mrow] = acc[r];
    }
  }
}

// ---------------------------------------------------------------------------
// 5) OI[i, h*64+d] = (attn[h] @ V[h])[i,d] + (WEsum[h] @ We_h)[i,d]
//    (V already holds +bv+be; two K-loops accumulate into one WMMA tile)
// ---------------------------------------------------------------------------
__global__ __launch_bounds__(32) void k_out(
    const float* __restrict__ ATTN, const float* __restrict__ V,
    const float* __restrict__ WEsum, const float* __restrict__ We,
    float* __restrict__ OI)
{
  const int lane = threadIdx.x & 31;
  const int mrow = lane & 15, khalf = lane >> 4;
  const int d0 = blockIdx.x * 16, i0 = blockIdx.y * 16, h = blockIdx.z;

  v8f acc = {};
  for (int kk = 0; kk < N_; kk += 4) {          // attn @ V  (K = 512 over j)
    const int ka = kk + 2 * khalf;
    v2f a, b;
    a.x = ATTN[((size_t)h * N_ + i0 + mrow) * N_ + ka];
    a.y = ATTN[((size_t)h * N_ + i0 + mrow) * N_ + ka + 1];
    b.x = V[(h * N_ + ka) * DH_ + d0 + mrow];
    b.y = V[(h * N_ + ka + 1) * DH_ + d0 + mrow];
    acc = wmma4(a, b, acc);
  }
  for (int kk = 0; kk < DE_; kk += 4) {         // + WEsum @ We_h (K = 64 over c)
    const int ka = kk + 2 * khalf;
    v2f a, b;
    a.x = WEsum[((size_t)h * N_ + i0 + mrow) * DE_ + ka];
    a.y = WEsum[((size_t)h * N_ + i0 + mrow) * DE_ + ka + 1];
    b.x = We[ka * INNER_ + h * DH_ + d0 + mrow];
    b.y = We[(ka + 1) * INNER_ + h * DH_ + d0 + mrow];
    acc = wmma4(a, b, acc);
  }
  for (int r = 0; r < 8; ++r) {
    const int n = i0 + r + 8 * khalf;
    OI[n * INNER_ + h * DH_ + d0 + mrow] = acc[r];  // (b, n, h*dh) layout
  }
}

// ---------------------------------------------------------------------------
// 6) out = OI @ Wo + bo    (512x512 @ 512x128)
// ---------------------------------------------------------------------------
__global__ __launch_bounds__(32) void k_final(
    const float* __restrict__ OI, const float* __restrict__ Wo,
    const float* __restrict__ bo, float* __restrict__ out)
{
  const int lane = threadIdx.x & 31;
  const int mrow = lane & 15, khalf = lane >> 4;
  const int c0 = blockIdx.x * 16, i0 = blockIdx.y * 16;

  v8f acc = {};
  for (int kk = 0; kk < INNER_; kk += 4) {
    const int ka = kk + 2 * khalf;
    v2f a, b;
    a.x = OI[(i0 + mrow) * INNER_ + ka];
    a.y = OI[(i0 + mrow) * INNER_ + ka + 1];
    b.x = Wo[ka * DNODE + c0 + mrow];
    b.y = Wo[(ka + 1) * DNODE + c0 + mrow];
    acc = wmma4(a, b, acc);
  }
  const float bias = bo[c0 + mrow];
  for (int r = 0; r < 8; ++r)
    out[(i0 + r + 8 * khalf) * DNODE + c0 + mrow] = acc[r] + bias;
}

// ---------------------------------------------------------------------------
extern "C" void kernel_launch(void* const* d_in, const int* in_sizes, int n_in,
                              void* d_out, int out_size, void* d_ws, size_t ws_size,
                              hipStream_t stream) {
  const float* nodes = (const float*)d_in[0];
  const float* edges = (const float*)d_in[1];
  const unsigned char* mask = (const unsigned char*)d_in[2];
  const float* Wq = (const float*)d_in[3];  const float* bq = (const float*)d_in[4];
  const float* Wk = (const float*)d_in[5];  const float* bk = (const float*)d_in[6];
  const float* Wv = (const float*)d_in[7];  const float* bv = (const float*)d_in[8];
  const float* We = (const float*)d_in[9];  const float* be = (const float*)d_in[10];
  const float* Wo = (const float*)d_in[11]; const float* bo = (const float*)d_in[12];
  float* out = (float*)d_out;
  float* ws  = (float*)d_ws;

  // Workspace layout (floats), total ~22 MB:
  float* Q    = ws;                              // H*N*DH
  float* K    = Q    + H_ * N_ * DH_;            // H*N*DH
  float* V    = K    + H_ * N_ * DH_;            // H*N*DH
  float* QE   = V    + H_ * N_ * DH_;            // H*N*DE
  float* SQK  = QE   + H_ * N_ * DE_;            // H*N*N
  float* ATTN = SQK  + (size_t)H_ * N_ * N_;     // H*N*N
  float* WEs  = ATTN + (size_t)H_ * N_ * N_;     // H*N*DE
  float* OI   = WEs  + H_ * N_ * DE_;            // N*INNER

  k_qkv<<<dim3(INNER_ / 16, N_ / 16, 3), 32, 0, stream>>>(
      nodes, Wq, bq, Wk, bk, Wv, bv, be, Q, K, V);
  k_qe<<<dim3(DE_ / 16, N_ / 16, H_), 32, 0, stream>>>(Q, We, QE);
  k_sqk<<<dim3(N_ / 16, N_ / 16, H_), 32, 0, stream>>>(Q, K, SQK);

  const size_t shbytes = (size_t)(N_ * EDS + H_ * SS + H_ * DE_) * sizeof(float);
  k_attn<<<N_, 256, shbytes, stream>>>(edges, mask, QE, SQK, ATTN, WEs);

  k_out<<<dim3(DH_ / 16, N_ / 16, H_), 32, 0, stream>>>(ATTN, V, WEs, We, OI);
  k_final<<<dim3(DNODE / 16, N_ / 16), 32, 0, stream>>>(OI, Wo, bo, out);
}